// GAT_4260607558248
// MI455X (gfx1250) — compile-verified
//
#include <hip/hip_runtime.h>

#define NN   8192
#define DDIM 512
#define HH   256
#define NCLS 16
#define AW   (NN / 32)   // 256 adjacency words per row

typedef float    v2f  __attribute__((ext_vector_type(2)));
typedef float    v8f  __attribute__((ext_vector_type(8)));
typedef unsigned u32x4 __attribute__((ext_vector_type(4)));
typedef int      i32x4 __attribute__((ext_vector_type(4)));
typedef int      i32x8 __attribute__((ext_vector_type(8)));

#if defined(__has_builtin)
#if __has_builtin(__builtin_amdgcn_tensor_load_to_lds) && \
    __has_builtin(__builtin_amdgcn_s_wait_tensorcnt)
#define TDM_OK 1
#endif
#endif

// ---------------------------------------------------------------------------
// TDM 2-D tile load: global -> LDS, data_size = 4 bytes.
// D# packing per CDNA5 ISA (groups 0/1; 2/3 zero for 2D).
// Dims/strides in 4-byte units. tile_d1 == 0 => 1-D tile.
// This toolchain exposes the 6-arg builtin (g0:4dw, g1:8dw, g2:4dw, g3:4dw,
// extra 8dw group, cpol) -- zero-fill the unused trailing group.
// ---------------------------------------------------------------------------
#ifdef TDM_OK
__device__ __forceinline__ void tdm_load_2d_b32(unsigned lds_off, const void* gptr,
                                                unsigned tensor_d0, unsigned tensor_d1,
                                                unsigned tile_d0, unsigned tile_d1,
                                                unsigned stride0) {
    unsigned long long ga = (unsigned long long)gptr;
    u32x4 g0;
    i32x8 g1;
    i32x4 g2 = {};
    i32x4 g3 = {};
    i32x8 g4 = {};
    g0[0] = 1u;                                           // count=1, user-mode
    g0[1] = lds_off;                                      // LDS byte address
    g0[2] = (unsigned)(ga & 0xFFFFFFFFu);                 // global_addr[31:0]
    g0[3] = (unsigned)((ga >> 32) & 0x01FFFFFFu)          // global_addr[56:32]
          | (2u << 30);                                   // type = 2 ("image")
    g1[0] = (int)(2u << 16);                              // data_size = 2 (4B)
    g1[1] = (int)((tensor_d0 & 0xFFFFu) << 16);           // tensor_dim0[15:0]
    g1[2] = (int)(((tensor_d0 >> 16) & 0xFFFFu)           // tensor_dim0[31:16]
          | ((tensor_d1 & 0xFFFFu) << 16));               // tensor_dim1[15:0]
    g1[3] = (int)(((tensor_d1 >> 16) & 0xFFFFu)           // tensor_dim1[31:16]
          | ((tile_d0 & 0xFFFFu) << 16));                 // tile_dim0
    g1[4] = (int)(tile_d1 & 0xFFFFu);                     // tile_dim1, tile_dim2=0
    g1[5] = (int)stride0;                                 // tensor_dim0_stride[31:0]
    g1[6] = 0;                                            // stride0 hi / stride1 lo
    g1[7] = 0;                                            // stride1 hi
    __builtin_amdgcn_tensor_load_to_lds(g0, g1, g2, g3, g4, 0);
}
#endif

// ---------------------------------------------------------------------------
// Kernel 1: h = x @ W^T   [8192x512] x [512x256] -> [8192x256], f32 WMMA.
// One wave computes one 16x16 output tile; K loop in steps of 4.
// ---------------------------------------------------------------------------
__global__ __launch_bounds__(256) void gat_h_gemm(const float* __restrict__ x,
                                                  const float* __restrict__ W,
                                                  float* __restrict__ h) {
    const int lane = threadIdx.x & 31;
    const int wid  = threadIdx.x >> 5;
    const int tile = blockIdx.x * 8 + wid;   // (NN/16)*(HH/16) = 8192 tiles
    const int rowt = tile >> 4;              // 512 row tiles
    const int colt = tile & 15;              // 16 col tiles
    const int m  = lane & 15;
    const int hi = lane >> 4;

    const float* xrow = x + (size_t)(rowt * 16 + m) * DDIM; // A: row m
    const float* wrow = W + (size_t)(colt * 16 + m) * DDIM; // B: n = m

    v8f acc = {};
#pragma unroll 4
    for (int k = 0; k < DDIM; k += 4) {
        v2f a, b;
        a.x = xrow[k + 2 * hi];
        a.y = xrow[k + 2 * hi + 1];
        b.x = wrow[k + 2 * hi];
        b.y = wrow[k + 2 * hi + 1];
        acc = __builtin_amdgcn_wmma_f32_16x16x4_f32(
            false, a, false, b, (short)0, acc, false, false);
    }
#pragma unroll
    for (int r = 0; r < 8; ++r)
        h[(size_t)(rowt * 16 + r + 8 * hi) * HH + colt * 16 + m] = acc[r];
}

// ---------------------------------------------------------------------------
// Kernel 2: f1 = h@a1, f2 = h@a2.  One wave per row (256-elem dot).
// ---------------------------------------------------------------------------
__global__ __launch_bounds__(256) void gat_f1f2(const float* __restrict__ h,
                                                const float* __restrict__ a1,
                                                const float* __restrict__ a2,
                                                float* __restrict__ f1,
                                                float* __restrict__ f2) {
    const int lane = threadIdx.x & 31;
    const int wid  = threadIdx.x >> 5;
    const int row  = blockIdx.x * 8 + wid;
    const float* hr = h + (size_t)row * HH;
    float s1 = 0.f, s2 = 0.f;
#pragma unroll
    for (int k = lane; k < HH; k += 32) {
        float v = hr[k];
        s1 += v * a1[k];
        s2 += v * a2[k];
    }
#pragma unroll
    for (int off = 16; off; off >>= 1) {
        s1 += __shfl_xor(s1, off);
        s2 += __shfl_xor(s2, off);
    }
    if (lane == 0) { f1[row] = s1; f2[row] = s2; }
}

// ---------------------------------------------------------------------------
// Kernel 3a: zero the adjacency bitmask (re-zero every call: atomicOr
// accumulates and the harness replays the graph).
// ---------------------------------------------------------------------------
__global__ void gat_zero(unsigned* __restrict__ adj, int nwords) {
    int i = blockIdx.x * blockDim.x + threadIdx.x;
    if (i < nwords) adj[i] = 0u;
}

// ---------------------------------------------------------------------------
// Kernel 3b: scatter edges + self-loops into the bitmask.
// ---------------------------------------------------------------------------
__global__ void gat_edges(const int* __restrict__ ei, int n_edges,
                          unsigned* __restrict__ adj) {
    int t = blockIdx.x * blockDim.x + threadIdx.x;
    if (t < n_edges) {
        int r = ei[t];
        int c = ei[n_edges + t];
        if ((unsigned)r < NN && (unsigned)c < NN)
            atomicOr(adj + (size_t)r * AW + (c >> 5), 1u << (c & 31));
    } else if (t < n_edges + NN) {
        int i = t - n_edges;
        atomicOr(adj + (size_t)i * AW + (i >> 5), 1u << (i & 31));
    }
}

// ---------------------------------------------------------------------------
// Kernel 4: fused flash-style GAT row-block.
//   16 rows/block; online softmax over 8192 cols in 16-wide chunks;
//   chunk-level AND K=4 sub-level adjacency skip (exact: masked cols give
//   p=0); P@h with f32 WMMA; fused elu + fc writes logits directly.
//   Block = 256 threads = 8 waves; wave w owns output columns [32w, 32w+32).
// ---------------------------------------------------------------------------
__global__ __launch_bounds__(256) void gat_attn(const float* __restrict__ h,
                                                const float* __restrict__ f1g,
                                                const float* __restrict__ f2g,
                                                const unsigned* __restrict__ adj,
                                                const float* __restrict__ fc_w,
                                                const float* __restrict__ fc_b,
                                                float* __restrict__ logits) {
    __shared__ float    f2s[NN];        // 32 KB: all f2
    __shared__ unsigned adjs[16 * AW];  // 16 KB: this block's adjacency rows

    const int tid     = threadIdx.x;
    const int lane    = tid & 31;
    const int wid     = tid >> 5;
    const int rowbase = blockIdx.x * 16;

#ifdef TDM_OK
    if (wid == 0) {
        // Two TDM descriptors: 1-D f2 (8192 x f32) and 2-D adjacency tile
        // (256 words x 16 rows, row stride 256 words). EXEC-independent DMA.
        tdm_load_2d_b32((unsigned)(size_t)&f2s[0], f2g,
                        NN, 1, NN, 0, NN);
        tdm_load_2d_b32((unsigned)(size_t)&adjs[0], adj + (size_t)rowbase * AW,
                        AW, NN, AW, 16, AW);
        __builtin_amdgcn_s_wait_tensorcnt(0);
    }
#else
    for (int i = tid; i < NN; i += 256) f2s[i] = f2g[i];
    for (int i = tid; i < 16 * AW; i += 256)
        adjs[i] = adj[(size_t)(rowbase + (i >> 8)) * AW + (i & (AW - 1))];
#endif
    __syncthreads();

    const int   m    = lane & 15;
    const int   hi   = lane >> 4;
    const float f1v  = f1g[rowbase + m];
    const int   col0 = wid * 32;
    const unsigned* myadj = adjs + m * AW;

    float mrow = -INFINITY, lrow = 0.f;   // per-row online softmax state
    v8f acc0 = {}, acc1 = {};             // 16x32 accumulator per wave

    for (int jb = 0; jb < NN; jb += 16) {
        const unsigned w    = myadj[jb >> 5];
        const unsigned bits = (w >> (jb & 16)) & 0xFFFFu;
        if (__ballot(bits != 0u) == 0ull) continue;   // empty 16x16 chunk

        // e-values, computed directly in WMMA A-operand positions:
        // value (s,t) is column j = jb + 4s + 2*hi + t of row m.
        float ev[4][2];
        float cmax = -INFINITY;
#pragma unroll
        for (int s = 0; s < 4; ++s)
#pragma unroll
            for (int t = 0; t < 2; ++t) {
                const int off = 4 * s + 2 * hi + t;
                float e = f1v + f2s[jb + off];
                e = e > 0.f ? e : 0.01f * e;                    // leaky_relu
                if (!((bits >> off) & 1u) || e == 0.f)          // mask (incl. exact-0 quirk)
                    e = -INFINITY;
                ev[s][t] = e;
                cmax = fmaxf(cmax, e);
            }
        cmax = fmaxf(cmax, __shfl_xor(cmax, 16));   // full 16-col row max
        const float mnew  = fmaxf(mrow, cmax);
        const float scale = (mnew == mrow) ? 1.f : __expf(mrow - mnew);

        float p[4][2];
        float psum = 0.f;
#pragma unroll
        for (int s = 0; s < 4; ++s)
#pragma unroll
            for (int t = 0; t < 2; ++t) {
                float pv = (ev[s][t] == -INFINITY) ? 0.f : __expf(ev[s][t] - mnew);
                p[s][t] = pv;
                psum += pv;
            }
        psum += __shfl_xor(psum, 16);
        lrow = lrow * scale + psum;
        mrow = mnew;

        // Rescale accumulators: D-tile row for vgpr r is (r + 8*hi);
        // lane (r+8*hi) holds that row's scale -> broadcast via bpermute.
#pragma unroll
        for (int r = 0; r < 8; ++r) {
            const float sc = __shfl(scale, r + 8 * hi);
            acc0[r] *= sc;
            acc1[r] *= sc;
        }

        // P[16x16] @ h[jb:jb+16, col0:col0+32] via up to 8x WMMA f32 16x16x4,
        // with K=4 sub-chunk skip (sub empty with p~77% at 0.40% density).
#pragma unroll
        for (int s = 0; s < 4; ++s) {
            if (__ballot(((bits >> (4 * s)) & 0xFu) != 0u) == 0ull) continue;
            v2f a; a.x = p[s][0]; a.y = p[s][1];
            const float* hb = h + (size_t)(jb + 4 * s + 2 * hi) * HH;
            v2f b0, b1;
            b0.x = hb[col0 + m];            // K = 4s + 2hi
            b0.y = hb[HH + col0 + m];       // K = 4s + 2hi + 1
            b1.x = hb[col0 + 16 + m];
            b1.y = hb[HH + col0 + 16 + m];
            acc0 = __builtin_amdgcn_wmma_f32_16x16x4_f32(
                false, a, false, b0, (short)0, acc0, false, false);
            acc1 = __builtin_amdgcn_wmma_f32_16x16x4_f32(
                false, a, false, b1, (short)0, acc1, false, false);
        }
    }

    // Finalize: divide by softmax denom, elu, stage 16x256 tile in LDS.
    __syncthreads();
    float* outstage = f2s;   // reuse LDS (f2/adj dead); row stride 257
#pragma unroll
    for (int r = 0; r < 8; ++r) {
        const int   orow = r + 8 * hi;
        const float ls   = __shfl(lrow, orow);
        float v0 = acc0[r] / ls;
        float v1 = acc1[r] / ls;
        v0 = v0 > 0.f ? v0 : __expf(v0) - 1.f;   // elu
        v1 = v1 > 0.f ? v1 : __expf(v1) - 1.f;
        outstage[orow * 257 + col0 + m]      = v0;
        outstage[orow * 257 + col0 + 16 + m] = v1;
    }
    __syncthreads();

    // fc: 256 threads = 16 rows x 16 classes, each a 256-elem dot.
    const int orow = tid >> 4, cls = tid & 15;
    const float* fw = fc_w + cls * HH;
    const float* os = outstage + orow * 257;
    float s = fc_b[cls];
#pragma unroll 8
    for (int k = 0; k < HH; ++k) s += os[k] * fw[k];
    logits[(size_t)(rowbase + orow) * NCLS + cls] = s;
}

// ---------------------------------------------------------------------------
extern "C" void kernel_launch(void* const* d_in, const int* in_sizes, int n_in,
                              void* d_out, int out_size, void* d_ws, size_t ws_size,
                              hipStream_t stream) {
    const float* x  = (const float*)d_in[0];
    const int*   ei = (const int*)d_in[1];     // int32 (jax default x64-off)
    const float* W  = (const float*)d_in[2];
    const float* a1 = (const float*)d_in[3];
    const float* a2 = (const float*)d_in[4];
    const float* fw = (const float*)d_in[5];
    const float* fb = (const float*)d_in[6];
    float* logits = (float*)d_out;

    char* ws = (char*)d_ws;
    float*    h   = (float*)ws;                                  // 8 MB
    float*    f1  = (float*)(ws + (size_t)8 * 1024 * 1024);      // 32 KB
    float*    f2  = (float*)(ws + (size_t)8 * 1024 * 1024 + 32 * 1024);
    unsigned* adj = (unsigned*)(ws + (size_t)8 * 1024 * 1024 + 64 * 1024); // 8 MB

    const int n_edges = in_sizes[1] / 2;
    const int nwords  = NN * AW;

    gat_h_gemm<<<(NN / 16) * (HH / 16) / 8, 256, 0, stream>>>(x, W, h);
    gat_f1f2<<<NN / 8, 256, 0, stream>>>(h, a1, a2, f1, f2);
    gat_zero<<<(nwords + 255) / 256, 256, 0, stream>>>(adj, nwords);
    gat_edges<<<(n_edges + NN + 255) / 256, 256, 0, stream>>>(ei, n_edges, adj);
    gat_attn<<<NN / 16, 256, 0, stream>>>(h, f1, f2, adj, fw, fb, logits);
}